// MHA_2688649527670
// MI455X (gfx1250) — compile-verified
//
#include <hip/hip_runtime.h>
#include <math.h>

#define MODEL_DIM 128
#define NUM_HEADS 8
#define BATCH 4
#define SEQ 2048
#define GSPLIT 8          // split-K factor for the Gram matrices

typedef __attribute__((ext_vector_type(2))) float v2f;
typedef __attribute__((ext_vector_type(8))) float v8f;

// D = A(16x4,f32) * B(4x16,f32) + C(16x16,f32); wave32, EXEC all-ones.
__device__ __forceinline__ v8f wmma4(v2f a, v2f b, v8f c) {
  return __builtin_amdgcn_wmma_f32_16x16x4_f32(
      /*neg_a=*/false, a, /*neg_b=*/false, b,
      /*c_mod=*/(short)0, c, /*reuse_a=*/false, /*reuse_b=*/false);
}

// ---------------------------------------------------------------------------
// Kernel 1a: partial Grams. grid = (BATCH, GSPLIT); each block covers
// SEQ/GSPLIT = 256 rows of x_b and emits a full 128x128 partial G.
// 8 waves, 16-row strip each; 32-row x chunks staged in LDS (16 KB).
// ---------------------------------------------------------------------------
__global__ void __launch_bounds__(256) gram_partial_kernel(const float* __restrict__ x,
                                                           float* __restrict__ Gpart) {
  __shared__ float xs[32 * 128];
  const int b     = blockIdx.x;
  const int split = blockIdx.y;
  const int tid   = threadIdx.x;
  const int lane  = tid & 31;
  const int mBase = (tid >> 5) * 16;
  const int lm    = lane & 15;
  const int khalf = (lane >> 4) << 1;   // lanes 16-31 hold K+2,K+3
  const float* xb = x + (size_t)b * SEQ * MODEL_DIM
                      + (size_t)split * (SEQ / GSPLIT) * MODEL_DIM;

  v8f zero = {0.f, 0.f, 0.f, 0.f, 0.f, 0.f, 0.f, 0.f};
  v8f acc[8];
#pragma unroll
  for (int j = 0; j < 8; ++j) acc[j] = zero;

  for (int c = 0; c < (SEQ / GSPLIT) / 32; ++c) {   // 8 chunks of 32 rows
    for (int i = tid; i < 1024; i += 256) {
      int row = i >> 5, c4 = (i & 31) << 2;
      *(float4*)&xs[row * 128 + c4] =
          *(const float4*)&xb[(size_t)(c * 32 + row) * 128 + c4];
    }
    __syncthreads();
#pragma unroll
    for (int k0 = 0; k0 < 32; k0 += 4) {
      const int kk = k0 + khalf;
      v2f a;                              // A[m,k] = x[k, mBase+m]  (x^T)
      a.x = xs[kk * 128 + mBase + lm];
      a.y = xs[(kk + 1) * 128 + mBase + lm];
#pragma unroll
      for (int j = 0; j < 8; ++j) {       // B[k,n] = x[k, n]
        v2f bb;
        bb.x = xs[kk * 128 + j * 16 + lm];
        bb.y = xs[(kk + 1) * 128 + j * 16 + lm];
        acc[j] = wmma4(a, bb, acc[j]);
      }
    }
    __syncthreads();
  }

  float* g = Gpart + (size_t)(b * GSPLIT + split) * 16384;
  const int rowOff = (lane & 16) ? 8 : 0; // C layout: lanes 16-31 -> M+8
#pragma unroll
  for (int j = 0; j < 8; ++j)
#pragma unroll
    for (int r = 0; r < 8; ++r)
      g[(mBase + r + rowOff) * 128 + (j * 16 + lm)] = acc[j][r];
}

// Kernel 1b: deterministic tree-less fixed-order reduction of the partials.
__global__ void __launch_bounds__(256) gram_reduce_kernel(const float* __restrict__ Gpart,
                                                          float* __restrict__ G) {
  const int b = blockIdx.x;
  for (int e = threadIdx.x; e < 16384; e += 256) {
    float s = 0.f;
#pragma unroll
    for (int p = 0; p < GSPLIT; ++p)
      s += Gpart[(size_t)(b * GSPLIT + p) * 16384 + e];
    G[(size_t)b * 16384 + e] = s;
  }
}

// ---------------------------------------------------------------------------
// 128x(64)x128 fp32 GEMM strip helper: 8 waves/block, 16 rows per wave,
// 64 output columns starting at nBase (grid.y selects the half).
// tA: A[m,k] = A[k*128+m]; tB: B[k,n] = B[n*128+k]
// ---------------------------------------------------------------------------
__device__ __forceinline__ void gemm128_half(const float* __restrict__ A, int tA,
                                             const float* __restrict__ B, int tB,
                                             float* __restrict__ C, float scale,
                                             int nBase) {
  const int tid   = threadIdx.x;
  const int lane  = tid & 31;
  const int mBase = (tid >> 5) * 16, lm = lane & 15, khalf = (lane >> 4) << 1;

  v8f zero = {0.f, 0.f, 0.f, 0.f, 0.f, 0.f, 0.f, 0.f};
  v8f acc[4];
#pragma unroll
  for (int j = 0; j < 4; ++j) acc[j] = zero;

  const int m = mBase + lm;
  for (int k0 = 0; k0 < 128; k0 += 4) {
    const int kk = k0 + khalf;
    v2f a;
    if (tA) { a.x = A[kk * 128 + m];  a.y = A[(kk + 1) * 128 + m]; }
    else    { a.x = A[m * 128 + kk];  a.y = A[m * 128 + kk + 1];   }
#pragma unroll
    for (int j = 0; j < 4; ++j) {
      const int n = nBase + j * 16 + lm;
      v2f bb;
      if (tB) { bb.x = B[n * 128 + kk];  bb.y = B[n * 128 + kk + 1]; }
      else    { bb.x = B[kk * 128 + n];  bb.y = B[(kk + 1) * 128 + n]; }
      acc[j] = wmma4(a, bb, acc[j]);
    }
  }
  const int rowOff = (lane & 16) ? 8 : 0;
#pragma unroll
  for (int j = 0; j < 4; ++j)
#pragma unroll
    for (int r = 0; r < 8; ++r)
      C[(mBase + r + rowOff) * 128 + nBase + j * 16 + lm] = acc[j][r] * scale;
}

// N-chain stages: T1 = G*Wv^T ; T2 = Wk*T1 ; N = Wq^T*T2 * (1/sqrt(D))
__global__ void __launch_bounds__(256) stage1_kernel(const float* __restrict__ G,
                                                     const float* __restrict__ Wv,
                                                     float* __restrict__ T1) {
  const int bh = blockIdx.x, b = bh >> 3, h = bh & 7;
  gemm128_half(G + b * 16384, 0, Wv + h * 16384, 1, T1 + bh * 16384, 1.0f,
               blockIdx.y * 64);
}
__global__ void __launch_bounds__(256) stage2_kernel(const float* __restrict__ Wk,
                                                     const float* __restrict__ T1,
                                                     float* __restrict__ T2) {
  const int bh = blockIdx.x, h = bh & 7;
  gemm128_half(Wk + h * 16384, 0, T1 + bh * 16384, 0, T2 + bh * 16384, 1.0f,
               blockIdx.y * 64);
}
__global__ void __launch_bounds__(256) stage3_kernel(const float* __restrict__ Wq,
                                                     const float* __restrict__ T2,
                                                     float* __restrict__ Nout) {
  const int bh = blockIdx.x, h = bh & 7;
  gemm128_half(Wq + h * 16384, 1, T2 + bh * 16384, 0, Nout + bh * 16384,
               0.08838834764831845f /* 1/sqrt(128) */, blockIdx.y * 64);
}

// ---------------------------------------------------------------------------
// Kernel 3 (dominant, BW-bound): Z[b,h] = softmax_rows(x_b * N[b,h])
// grid = (S/64 tiles, B*H); 4 waves/block, 16 output rows per wave.
// x tile (64x132, padded vs bank conflicts) AND N (128x128) staged in LDS
// (~97 KB of the WGP's 320 KB) so the whole inner loop runs on ds_load_2addr.
// ---------------------------------------------------------------------------
__global__ void __launch_bounds__(128) attn_out_kernel(const float* __restrict__ x,
                                                       const float* __restrict__ Nmat,
                                                       float* __restrict__ Z) {
  __shared__ float xs[64 * 132];
  __shared__ float ns[128 * 128];
  const int sTile = blockIdx.x;      // 0..31
  const int bh    = blockIdx.y;      // 0..31
  const int b     = bh >> 3;
  const int tid   = threadIdx.x;
  const int lane  = tid & 31;
  const int lm    = lane & 15, khalf = (lane >> 4) << 1;
  const int mRow  = (tid >> 5) * 16;
  const int sBase = sTile * 64;
  const float* xb = x + (size_t)b * SEQ * MODEL_DIM;
  const float* Np = Nmat + (size_t)bh * 16384;

  for (int i = tid; i < 2048; i += 128) {          // x tile: 64 rows
    int row = i >> 5, c4 = (i & 31) << 2;
    *(float4*)&xs[row * 132 + c4] =
        *(const float4*)&xb[(size_t)(sBase + row) * 128 + c4];
  }
  for (int i = tid; i < 4096; i += 128) {          // N: 128x128
    *(float4*)&ns[i << 2] = *(const float4*)&Np[i << 2];
  }
  __syncthreads();

  v8f zero = {0.f, 0.f, 0.f, 0.f, 0.f, 0.f, 0.f, 0.f};
  v8f acc[8];
#pragma unroll
  for (int j = 0; j < 8; ++j) acc[j] = zero;

#pragma unroll 4
  for (int k0 = 0; k0 < 128; k0 += 4) {
    const int kk = k0 + khalf;
    v2f a;                                // A[m,k] = x[sBase+mRow+m, k]
    a.x = xs[(mRow + lm) * 132 + kk];
    a.y = xs[(mRow + lm) * 132 + kk + 1];
#pragma unroll
    for (int j = 0; j < 8; ++j) {         // B[k,n] = N[k, n] from LDS
      const int n = j * 16 + lm;
      v2f bb;
      bb.x = ns[kk * 128 + n];
      bb.y = ns[(kk + 1) * 128 + n];
      acc[j] = wmma4(a, bb, acc[j]);
    }
  }

  // Row softmax over the 128 output columns. A row of Y lives in acc[0..7][r]
  // across one 16-lane half; xor masks 1/2/4/8 stay within each half.
  const int rowOff = (lane & 16) ? 8 : 0;
  float* zp = Z + (size_t)bh * SEQ * MODEL_DIM;
#pragma unroll
  for (int r = 0; r < 8; ++r) {
    float m = -1e30f;
#pragma unroll
    for (int j = 0; j < 8; ++j) m = fmaxf(m, acc[j][r]);
    m = fmaxf(m, __shfl_xor(m, 1, 32));
    m = fmaxf(m, __shfl_xor(m, 2, 32));
    m = fmaxf(m, __shfl_xor(m, 4, 32));
    m = fmaxf(m, __shfl_xor(m, 8, 32));
    float e[8], s = 0.f;
#pragma unroll
    for (int j = 0; j < 8; ++j) { e[j] = __expf(acc[j][r] - m); s += e[j]; }
    s += __shfl_xor(s, 1, 32);
    s += __shfl_xor(s, 2, 32);
    s += __shfl_xor(s, 4, 32);
    s += __shfl_xor(s, 8, 32);
    const float inv = 1.0f / s;
    const int srow = sBase + mRow + r + rowOff;
#pragma unroll
    for (int j = 0; j < 8; ++j)
      zp[(size_t)srow * 128 + j * 16 + lm] = e[j] * inv;
  }
}

// ---------------------------------------------------------------------------
extern "C" void kernel_launch(void* const* d_in, const int* in_sizes, int n_in,
                              void* d_out, int out_size, void* d_ws, size_t ws_size,
                              hipStream_t stream) {
  const float* x  = (const float*)d_in[0];
  const float* Wq = (const float*)d_in[1];
  const float* Wk = (const float*)d_in[2];
  const float* Wv = (const float*)d_in[3];
  float* Z  = (float*)d_out;
  float* ws = (float*)d_ws;

  // workspace layout (floats), ~6.5 MB total:
  //   Gpart[B*GSPLIT*16384] | G[B*16384] | T1[32*16384] | T2[32*16384]
  // T1 is reused to hold N after stage3.
  float* Gpart = ws;
  float* G     = Gpart + (size_t)BATCH * GSPLIT * 16384;
  float* T1    = G + (size_t)BATCH * 16384;
  float* T2    = T1 + (size_t)32 * 16384;

  gram_partial_kernel<<<dim3(BATCH, GSPLIT), dim3(256), 0, stream>>>(x, Gpart);
  gram_reduce_kernel <<<dim3(BATCH),         dim3(256), 0, stream>>>(Gpart, G);
  stage1_kernel<<<dim3(32, 2), dim3(256), 0, stream>>>(G,  Wv, T1);
  stage2_kernel<<<dim3(32, 2), dim3(256), 0, stream>>>(Wk, T1, T2);
  stage3_kernel<<<dim3(32, 2), dim3(256), 0, stream>>>(Wq, T2, T1); // N -> T1
  attn_out_kernel<<<dim3(32, 32), dim3(128), 0, stream>>>(x, T1, Z);
}